// NodeEmbedding_5583457485036
// MI455X (gfx1250) — compile-verified
//
#include <hip/hip_runtime.h>
#include <hip/hip_bf16.h>

typedef __attribute__((ext_vector_type(2))) float v2f;
typedef __attribute__((ext_vector_type(8))) float v8f;

#define N_NODES 40000
#define N_EDGES 640000
#define D 128
#define BN_EPS 1e-5f

// ---------------------------------------------------------------- init
__global__ void gcn_init(float* __restrict__ deg, float* __restrict__ stats, int n) {
    int i = blockIdx.x * blockDim.x + threadIdx.x;
    if (i < n) deg[i] = 1.0f;          // self-loop contributes 1 to every degree
    if (i < 256) stats[i] = 0.0f;      // gsum[128] + gsumsq[128]
}

// ---------------------------------------------------------------- degree
__global__ void gcn_degree(const int* __restrict__ dst, float* __restrict__ deg, int e) {
    int i = blockIdx.x * blockDim.x + threadIdx.x;
    if (i < e) atomicAdd(&deg[dst[i]], 1.0f);
}

__global__ void gcn_dinv(const float* __restrict__ deg, float* __restrict__ dinv, int n) {
    int i = blockIdx.x * blockDim.x + threadIdx.x;
    if (i < n) dinv[i] = rsqrtf(deg[i]);   // deg >= 1 always (self-loop)
}

// ---------------------------------------------------------------- GEMM h = x @ W  (f32 WMMA 16x16x4)
// One block = 16 rows of x, 8 waves each own a 16-col tile of W. K=128 in steps of 4.
// Epilogue stores h AND seeds out with the self-loop term h * dinv[row]^2.
__global__ __launch_bounds__(256) void gcn_gemm_wmma(const float* __restrict__ X,
                                                     const float* __restrict__ W,
                                                     const float* __restrict__ dinv,
                                                     float* __restrict__ H,
                                                     float* __restrict__ out) {
    const int lane  = threadIdx.x & 31;
    const int wave  = threadIdx.x >> 5;          // 0..7 -> output column tile
    const int row0  = blockIdx.x * 16;
    const int col0  = wave * 16;
    const int m     = lane & 15;                 // M (for A) / N (for B) index
    const int khalf = (lane >> 4) << 1;          // lanes 16-31 handle K+2,K+3

    const float* xrow = X + (row0 + m) * D;      // A: row m, two consecutive K elems
    v8f c = {};                                   // zero accumulator

    #pragma unroll 4
    for (int k = 0; k < D; k += 4) {
        v2f a, b;
        a.x = xrow[k + khalf];
        a.y = xrow[k + khalf + 1];
        const float* wrow = W + (k + khalf) * D + col0 + m;   // B: col m, two K rows
        b.x = wrow[0];
        b.y = wrow[D];
        c = __builtin_amdgcn_wmma_f32_16x16x4_f32(false, a, false, b,
                                                  (short)0, c, false, false);
    }

    // C/D layout: VGPR i -> row M=i (lanes 0-15) / M=i+8 (lanes 16-31), N=lane&15
    const int mbase = (lane >> 4) << 3;
    #pragma unroll
    for (int i = 0; i < 8; ++i) {
        const int row = row0 + mbase + i;
        const float hv = c[i];
        H[row * D + col0 + m] = hv;
        const float di = dinv[row];
        out[row * D + col0 + m] = hv * di * di;   // self-loop seed
    }
}

// ---------------------------------------------------------------- edge scatter: one wave per edge
__global__ __launch_bounds__(256) void gcn_scatter(const int* __restrict__ src,
                                                   const int* __restrict__ dst,
                                                   const float* __restrict__ H,
                                                   const float* __restrict__ dinv,
                                                   float* __restrict__ out, int e) {
    int edge = blockIdx.x * 8 + (threadIdx.x >> 5);
    if (edge >= e) return;
    int lane = threadIdx.x & 31;
    int s = src[edge];
    int t = dst[edge];
    float norm = dinv[s] * dinv[t];
    float4 hv = ((const float4*)(H + s * D))[lane];   // coalesced 512B gather per wave
    float* o = out + t * D + lane * 4;
    atomicAdd(o + 0, hv.x * norm);
    atomicAdd(o + 1, hv.y * norm);
    atomicAdd(o + 2, hv.z * norm);
    atomicAdd(o + 3, hv.w * norm);
}

// ---------------------------------------------------------------- per-channel stats of relu(agg + b)
// Read-only reduction pass: does NOT materialize the activated values.
__global__ __launch_bounds__(128) void gcn_relu_stats(const float* __restrict__ out,
                                                      const float* __restrict__ bias,
                                                      float* __restrict__ gsum,
                                                      float* __restrict__ gsumsq,
                                                      int rows_per_block, int n) {
    int d  = threadIdx.x;                 // channel
    int r0 = blockIdx.x * rows_per_block;
    int r1 = r0 + rows_per_block; if (r1 > n) r1 = n;
    float bv = bias[d];
    float s = 0.0f, ss = 0.0f;
    for (int r = r0; r < r1; ++r) {
        float v = out[r * D + d] + bv;
        v = v > 0.0f ? v : 0.0f;
        s  += v;
        ss += v * v;
    }
    atomicAdd(&gsum[d],   s);
    atomicAdd(&gsumsq[d], ss);
}

// ---------------------------------------------------------------- bias + relu + batchnorm (fused)
__global__ void gcn_bn(float* __restrict__ out, const float* __restrict__ bias,
                       const float* __restrict__ gsum, const float* __restrict__ gsumsq,
                       const float* __restrict__ gamma, const float* __restrict__ beta,
                       int total) {
    int i = blockIdx.x * blockDim.x + threadIdx.x;
    if (i >= total) return;
    int d = i & (D - 1);
    const float invN = 1.0f / (float)N_NODES;
    float mean = gsum[d] * invN;
    float var  = gsumsq[d] * invN - mean * mean;
    float v = out[i] + bias[d];
    v = v > 0.0f ? v : 0.0f;              // recompute relu(agg + b), saves a store pass
    out[i] = (v - mean) * rsqrtf(var + BN_EPS) * gamma[d] + beta[d];
}

// ---------------------------------------------------------------- launch
extern "C" void kernel_launch(void* const* d_in, const int* in_sizes, int n_in,
                              void* d_out, int out_size, void* d_ws, size_t ws_size,
                              hipStream_t stream) {
    const float* x     = (const float*)d_in[0];
    const int*   ei    = (const int*)  d_in[1];
    const float* W     = (const float*)d_in[2];
    const float* b     = (const float*)d_in[3];
    const float* gamma = (const float*)d_in[4];
    const float* beta  = (const float*)d_in[5];
    float* out = (float*)d_out;

    const int E = in_sizes[1] / 2;
    const int N = N_NODES;
    const int* src = ei;       // edge_index[0]
    const int* dst = ei + E;   // edge_index[1]

    // workspace layout
    char* ws = (char*)d_ws;
    float* h      = (float*)ws;                         // N*D f32 = 20,480,000 B
    float* deg    = (float*)(ws + (size_t)N * D * 4);   // N f32
    float* dinv   = deg + N;                            // N f32
    float* gsum   = dinv + N;                           // 128 f32
    float* gsumsq = gsum + D;                           // 128 f32

    gcn_init    <<<(N + 255) / 256, 256, 0, stream>>>(deg, gsum, N);
    gcn_degree  <<<(E + 255) / 256, 256, 0, stream>>>(dst, deg, E);
    gcn_dinv    <<<(N + 255) / 256, 256, 0, stream>>>(deg, dinv, N);

    // GEMM + fused self-loop seed (needs dinv ready)
    gcn_gemm_wmma<<<N / 16, 256, 0, stream>>>(x, W, dinv, h, out);

    gcn_scatter <<<(E + 7) / 8, 256, 0, stream>>>(src, dst, h, dinv, out, E);

    const int rows_per_block = 320;                     // 125 blocks
    gcn_relu_stats<<<(N + rows_per_block - 1) / rows_per_block, 128, 0, stream>>>(
        out, b, gsum, gsumsq, rows_per_block, N);

    const int total = N * D;
    gcn_bn<<<(total + 255) / 256, 256, 0, stream>>>(out, b, gsum, gsumsq, gamma, beta, total);
}